// time_attention_8658654069267
// MI455X (gfx1250) — compile-verified
//
#include <hip/hip_runtime.h>

typedef __attribute__((ext_vector_type(2))) float v2f;
typedef __attribute__((ext_vector_type(4))) float f4;
typedef __attribute__((ext_vector_type(8))) float v8f;

#define MULT    1024
#define HEADS   16
#define HEADDIM 64
#define ROWS    16                 // rows of N handled per block
#define LDA     1028               // padded LDS row stride in floats (bank-conflict free)
#define SMEM_FLOATS (4 * ROWS * LDA)
#define SMEM_BYTES  (SMEM_FLOATS * sizeof(float))

// Fully fused: parallel linears (WMMA f32 16x16x4) -> stride-16 softmax in LDS
// -> elementwise value merge -> output GEMM (WMMA) -> global store.
__global__ __launch_bounds__(256) void fused_time_attention_kernel(
    const float* __restrict__ t_inp, const float* __restrict__ lat_inp,
    const float* __restrict__ Wt, const float* __restrict__ bt,
    const float* __restrict__ Wl, const float* __restrict__ bl,
    const float* __restrict__ Wv, const float* __restrict__ bv,
    const float* __restrict__ Wo, const float* __restrict__ bo,
    float* __restrict__ out)
{
    extern __shared__ float smem[];
    float* sT = smem;                  // [ROWS x LDA] staged t rows
    float* sL = smem + 1 * ROWS * LDA; // [ROWS x LDA] staged lat rows
    float* sP = smem + 2 * ROWS * LDA; // [ROWS x LDA] t*lat products -> probs -> merged
    float* sV = smem + 3 * ROWS * LDA; // [ROWS x LDA] value embedding

    const int tid  = threadIdx.x;
    const int wave = tid >> 5;         // 8 waves (wave32)
    const int lane = tid & 31;
    const int row0 = blockIdx.x * ROWS;

    // ---- Stage 0: cooperative load of the 16-row input stripe into LDS ----
    for (int idx = tid; idx < ROWS * (MULT / 4); idx += 256) {
        int r  = idx >> 8;             // MULT/4 == 256 float4 per row
        int c4 = (idx & 255) << 2;
        f4 vt = *(const f4*)(t_inp   + (size_t)(row0 + r) * MULT + c4);
        f4 vl = *(const f4*)(lat_inp + (size_t)(row0 + r) * MULT + c4);
        *(f4*)(sT + r * LDA + c4) = vt;   // row stride 1028*4B keeps 16B alignment
        *(f4*)(sL + r * LDA + c4) = vl;
    }
    __syncthreads();

    // WMMA lane decomposition (f32 16x16x4): lanes 0-15 -> K={k,k+1}, 16-31 -> K={k+2,k+3}
    const int n    = lane & 15;        // N / M index within tile
    const int half = lane >> 4;
    const int koff = half * 2;

    // ---- Stage 1: three parallel linears; each wave owns 8 column tiles ----
    for (int j = wave * 8; j < wave * 8 + 8; ++j) {
        const int c = j * 16 + n;      // global output column 0..1023
        const int h = c & 15;          // reshape: col = o*16 + h
        const int o = c >> 4;
        const float* pBt = Wt + (size_t)(h * HEADDIM + o) * MULT;
        const float* pBl = Wl + (size_t)(h * HEADDIM + o) * MULT;
        const float* pBv = Wv + (size_t)(h * HEADDIM + o) * (2 * MULT);

        v8f at = {}, al = {}, av = {};
        #pragma unroll 4
        for (int k = 0; k < MULT; k += 4) {
            v2f aT = *(const v2f*)(sT + n * LDA + k + koff);
            v2f aL = *(const v2f*)(sL + n * LDA + k + koff);
            v2f bT = *(const v2f*)(pBt + k + koff);
            v2f bL = *(const v2f*)(pBl + k + koff);
            v2f bV = *(const v2f*)(pBv + k + koff);
            at = __builtin_amdgcn_wmma_f32_16x16x4_f32(false, aT, false, bT, (short)0, at, false, false);
            al = __builtin_amdgcn_wmma_f32_16x16x4_f32(false, aL, false, bL, (short)0, al, false, false);
            av = __builtin_amdgcn_wmma_f32_16x16x4_f32(false, aT, false, bV, (short)0, av, false, false);
        }
        #pragma unroll 4
        for (int k = 0; k < MULT; k += 4) {   // second half of concat(t,lat) for value path
            v2f aL = *(const v2f*)(sL + n * LDA + k + koff);
            v2f bV = *(const v2f*)(pBv + MULT + k + koff);
            av = __builtin_amdgcn_wmma_f32_16x16x4_f32(false, aL, false, bV, (short)0, av, false, false);
        }

        const float bt_v = bt[h * HEADDIM + o];
        const float bl_v = bl[h * HEADDIM + o];
        const float bv_v = bv[h * HEADDIM + o];
        #pragma unroll
        for (int r = 0; r < 8; ++r) {         // element r -> matrix row r + 8*half, col c
            float et = at[r] + bt_v;
            float el = al[r] + bl_v;
            float ev = av[r] + bv_v;
            sP[(r + 8 * half) * LDA + c] = et * el;
            sV[(r + 8 * half) * LDA + c] = ev;
        }
    }
    __syncthreads();

    // ---- Stage 2: softmax over HEAD dim (stride-16 columns), merge with value ----
    {
        const int r = tid >> 4;               // 0..15 rows
        const int h = tid & 15;               // 0..15 heads
        float* pr = sP + r * LDA + h;
        float* pv = sV + r * LDA + h;
        float mx = -3.402823466e38f;
        #pragma unroll
        for (int oo = 0; oo < HEADDIM; ++oo) mx = fmaxf(mx, pr[oo * 16]);
        float s = 0.f;
        #pragma unroll
        for (int oo = 0; oo < HEADDIM; ++oo) {
            float e = __expf(pr[oo * 16] - mx);
            pr[oo * 16] = e;
            s += e;
        }
        const float inv = 1.0f / s;
        #pragma unroll
        for (int oo = 0; oo < HEADDIM; ++oo)
            pr[oo * 16] = pr[oo * 16] * inv * pv[oo * 16];  // merged, in place
    }
    __syncthreads();

    // ---- Stage 3: out = merged @ Wo^T + bo (A from LDS, B rows of Wo from L2) ----
    for (int j = wave * 8; j < wave * 8 + 8; ++j) {
        const int c = j * 16 + n;
        const float* pWo = Wo + (size_t)c * MULT;   // B[k][c] = Wo[c*MULT + k]
        v8f acc = {};
        #pragma unroll 4
        for (int k = 0; k < MULT; k += 4) {
            v2f a = *(const v2f*)(sP + n * LDA + k + koff);
            v2f b = *(const v2f*)(pWo + k + koff);
            acc = __builtin_amdgcn_wmma_f32_16x16x4_f32(false, a, false, b, (short)0, acc, false, false);
        }
        const float bias = bo[c];
        #pragma unroll
        for (int r = 0; r < 8; ++r)
            out[(size_t)(row0 + r + 8 * half) * MULT + c] = acc[r] + bias;
    }
}

extern "C" void kernel_launch(void* const* d_in, const int* in_sizes, int n_in,
                              void* d_out, int out_size, void* d_ws, size_t ws_size,
                              hipStream_t stream) {
    (void)n_in; (void)out_size; (void)d_ws; (void)ws_size;
    const float* t_inp = (const float*)d_in[0];
    const float* lat   = (const float*)d_in[1];
    const float* Wt    = (const float*)d_in[2];
    const float* bt    = (const float*)d_in[3];
    const float* Wl    = (const float*)d_in[4];
    const float* bl    = (const float*)d_in[5];
    const float* Wv    = (const float*)d_in[6];
    const float* bv    = (const float*)d_in[7];
    const float* Wo    = (const float*)d_in[8];
    const float* bo    = (const float*)d_in[9];
    float* out = (float*)d_out;

    const int nrows = in_sizes[0] / MULT;   // 32768
    const int grid  = nrows / ROWS;         // 2048 blocks of 16 rows

    // 257 KB dynamic LDS (<= 320 KB/WGP on CDNA5); raise the per-kernel cap.
    hipFuncSetAttribute((const void*)fused_time_attention_kernel,
                        hipFuncAttributeMaxDynamicSharedMemorySize, (int)SMEM_BYTES);

    fused_time_attention_kernel<<<grid, 256, SMEM_BYTES, stream>>>(
        t_inp, lat, Wt, bt, Wl, bl, Wv, bv, Wo, bo, out);
}